// GATModel_77077483094939
// MI455X (gfx1250) — compile-verified
//
#include <hip/hip_runtime.h>
#include <hip/hip_bf16.h>

// ---------------------------------------------------------------------------
// GAT model on gfx1250 (MI455X). Dense projections use V_WMMA_F32_16X16X4_F32
// (exact f32 math on the matrix pipe) with an LDS-staged weight panel and a
// guard-free inner loop (K is a template constant; x is zero-padded to K=32).
// Edge softmax/scatter uses ordered-uint atomicMax + f32 atomicAdd.
// All kernels launched on `stream`, graph-capture safe.
// ---------------------------------------------------------------------------

#define N_NODES  50000
#define N_EDGES  800000
#define IN_F     30
#define IN_PAD   32
#define HEADS    8
#define DPH      32          // per-head dim
#define HD       (HEADS*DPH) // 256
#define GRAPHS   512
#define HID      128
#define ALPHA    0.2f
#define BN_EPS   1e-5f

#define M_TILES  (N_NODES / 16)   // 3125 exactly

typedef __attribute__((ext_vector_type(2))) float v2f;
typedef __attribute__((ext_vector_type(8))) float v8f;

// monotone float <-> uint ordering (for atomic max on floats)
__device__ __forceinline__ unsigned f2o(float f) {
    unsigned u = __float_as_uint(f);
    return (u & 0x80000000u) ? ~u : (u | 0x80000000u);
}
__device__ __forceinline__ float o2f(unsigned u) {
    unsigned b = (u & 0x80000000u) ? (u & 0x7FFFFFFFu) : ~u;
    return __uint_as_float(b);
}

// ------------------------------ zero fill ----------------------------------
__global__ void k_zero(unsigned* __restrict__ p, int n) {
    int t = blockIdx.x * blockDim.x + threadIdx.x;
    if (t < n) p[t] = 0u;
}

// ------------------- pad x [N,30] -> xp [N,32] (zero tail) -----------------
__global__ void k_pad(const float* __restrict__ x, float* __restrict__ xp, int n) {
    int t = blockIdx.x * blockDim.x + threadIdx.x;
    if (t >= n) return;
    int node = t >> 5, k = t & 31;
    xp[t] = (k < IN_F) ? x[node * IN_F + k] : 0.0f;
}

// --------------------- WMMA f32 GEMM: Out = A @ B (+bias) ------------------
// A: [M, K] row-major (K multiple of 4, fully valid), B: [Krows, 256],
// Out: [M, 256]. Block = 4 waves; each wave owns one 16x16 output tile
// against a shared LDS-staged B column-panel [K x 16]. No guards in the
// K-loop. Fragment layouts per CDNA5 ISA 7.12.2:
//   A 16x4 : row = lane&15;  K = vgpr + 2*(lane>>4)  (a.x/a.y K-consecutive)
//   B 4x16 : col = lane&15;  K = vgpr + 2*(lane>>4)
//   C/D    : col = lane&15;  row = vgpr + 8*(lane>>4)
template<int K>
__global__ __launch_bounds__(128) void k_gemm_f32(
    const float* __restrict__ A, const float* __restrict__ B,
    const float* __restrict__ bias, float* __restrict__ Out,
    int mtiles, int Krows)
{
    __shared__ float Bs[K * 16];
    const int tid     = threadIdx.x;       // 0..127
    const int colBase = blockIdx.y * 16;

    // one-time cooperative stage of B panel (guard only here, not in hot loop)
    for (int i = tid; i < K * 16; i += 128) {
        int krow = i >> 4;
        Bs[i] = (krow < Krows) ? B[(long)krow * HD + colBase + (i & 15)] : 0.0f;
    }
    __syncthreads();

    const int wave  = tid >> 5;
    const int mtile = blockIdx.x * 4 + wave;
    if (mtile >= mtiles) return;           // wave-uniform; EXEC stays all-1s
    const int lane = tid & 31;
    const int half = lane >> 4;
    const int l15  = lane & 15;
    const float* Arow = A + (long)(mtile * 16 + l15) * K;

    v8f acc = {};
#pragma unroll 8
    for (int k0 = 0; k0 < K; k0 += 4) {
        const int kA = k0 + 2 * half;
        v2f a = *(const v2f*)(Arow + kA);                    // 8B aligned load
        v2f b = { Bs[kA * 16 + l15], Bs[kA * 16 + 16 + l15] };
        acc = __builtin_amdgcn_wmma_f32_16x16x4_f32(
            /*neg_a=*/false, a, /*neg_b=*/false, b,
            /*c_mod=*/(short)0, acc, /*reuse_a=*/false, /*reuse_b=*/false);
    }

    const float bb = bias ? bias[colBase + l15] : 0.0f;
    float* OutT = Out + (long)(mtile * 16 + 8 * half) * HD + colBase + l15;
#pragma unroll
    for (int r = 0; r < 8; ++r)
        OutT[(long)r * HD] = acc[r] + bb;
}

// ----------------- attention logits el/er per (node, head) -----------------
__global__ void k_elr(const float* __restrict__ feat,
                      const float* __restrict__ al, const float* __restrict__ ar,
                      float* __restrict__ el, float* __restrict__ er, int n_nh)
{
    int t = blockIdx.x * blockDim.x + threadIdx.x;
    if (t >= n_nh) return;
    int n = t >> 3, h = t & 7;
    const float* f = feat + (long)n * HD + h * DPH;
    const float* a = al + h * DPH;
    const float* b = ar + h * DPH;
    float sl = 0.f, sr = 0.f;
#pragma unroll
    for (int d = 0; d < DPH; ++d) { sl += f[d] * a[d]; sr += f[d] * b[d]; }
    el[t] = sl; er[t] = sr;
}

// ------------- edge pass A: e = lrelu(el[src]+er[dst]); seg max ------------
__global__ void k_edge_max(const int* __restrict__ src, const int* __restrict__ dst,
                           const float* __restrict__ el, const float* __restrict__ er,
                           float* __restrict__ ebuf, unsigned* __restrict__ m_ord)
{
    int t = blockIdx.x * blockDim.x + threadIdx.x;
    if (t >= N_EDGES * HEADS) return;
    int e = t >> 3, h = t & 7;
    int sN = src[e], dN = dst[e];
    float ev = el[sN * HEADS + h] + er[dN * HEADS + h];
    ev = ev > 0.f ? ev : ALPHA * ev;                 // LeakyReLU
    ebuf[t] = ev;
    atomicMax(&m_ord[dN * HEADS + h], f2o(ev));
}

// ------------- edge pass B: ex = exp(e - m[dst]); seg sum ------------------
__global__ void k_edge_exp(const int* __restrict__ dst,
                           float* __restrict__ ebuf, const unsigned* __restrict__ m_ord,
                           float* __restrict__ s)
{
    int t = blockIdx.x * blockDim.x + threadIdx.x;
    if (t >= N_EDGES * HEADS) return;
    int e = t >> 3, h = t & 7;
    int dN = dst[e];
    float m = o2f(m_ord[dN * HEADS + h]);            // finite: this edge contributed
    float x = __expf(ebuf[t] - m);
    ebuf[t] = x;
    atomicAdd(&s[dN * HEADS + h], x);
}

// --------- edge pass C: agg[dst] += (ex/s[dst]) * feat[src] ----------------
// one block of 256 threads per edge -> coalesced feat read + atomic scatter
__global__ __launch_bounds__(256) void k_edge_msg(
    const int* __restrict__ src, const int* __restrict__ dst,
    const float* __restrict__ ebuf, const float* __restrict__ s,
    const float* __restrict__ feat, float* __restrict__ agg)
{
    int e = blockIdx.x;
    int j = threadIdx.x;          // h*32 + d
    int h = j >> 5;
    int sN = src[e], dN = dst[e];
    float a = ebuf[e * HEADS + h] / fmaxf(s[dN * HEADS + h], 1e-9f);
    atomicAdd(&agg[(long)dN * HD + j], a * feat[(long)sN * HD + j]);
}

// --------------------------- ELU (layer-1 finish) --------------------------
__global__ void k_elu(const float* __restrict__ in, float* __restrict__ out, int n) {
    int t = blockIdx.x * blockDim.x + threadIdx.x;
    if (t >= n) return;
    float v = in[t];
    out[t] = v > 0.f ? v : (__expf(v) - 1.0f);
}

// ------------------- agg2 init = h1 + b2 (identity residual) ---------------
__global__ void k_res2(const float* __restrict__ h1, const float* __restrict__ b2,
                       float* __restrict__ agg, int n) {
    int t = blockIdx.x * blockDim.x + threadIdx.x;
    if (t >= n) return;
    agg[t] = h1[t] + b2[t & (HD - 1)];
}

// -------- fused head-mean + sigmoid gate + per-graph readout ---------------
// one wave per node; lane = output feature d in [0,32)
__global__ __launch_bounds__(256) void k_readout(
    const float* __restrict__ agg, const int* __restrict__ gid,
    const float* __restrict__ Ww, const float* __restrict__ bw,
    float* __restrict__ gsum, unsigned* __restrict__ gmax_ord)
{
    int wave = threadIdx.x >> 5;
    int lane = threadIdx.x & 31;
    int n = blockIdx.x * 8 + wave;
    if (n >= N_NODES) return;
    float v = 0.f;
#pragma unroll
    for (int h = 0; h < HEADS; ++h) v += agg[(long)n * HD + h * DPH + lane];
    v *= (1.0f / HEADS);                             // h2[n, lane]
    float t = v * Ww[lane];
#pragma unroll
    for (int off = 16; off > 0; off >>= 1) t += __shfl_xor(t, off, 32);
    float w = 1.0f / (1.0f + __expf(-(t + bw[0])));  // sigmoid gate
    int g = gid[n];
    atomicAdd(&gsum[g * DPH + lane], w * v);
    atomicMax(&gmax_ord[g * DPH + lane], f2o(v));
}

// ------------------ MLP predictor: one block per graph ---------------------
__global__ __launch_bounds__(HID) void k_mlp(
    const float* __restrict__ gsum, const unsigned* __restrict__ gmax_ord,
    const float* __restrict__ Wp1, const float* __restrict__ bp1,
    const float* __restrict__ gamma, const float* __restrict__ beta,
    const float* __restrict__ rm, const float* __restrict__ rv,
    const float* __restrict__ Wp2, const float* __restrict__ bp2,
    float* __restrict__ out)
{
    __shared__ float gv[2 * DPH];
    __shared__ float red[HID];
    int g = blockIdx.x, j = threadIdx.x;
    if (j < DPH) gv[j] = gsum[g * DPH + j];
    else if (j < 2 * DPH) {
        unsigned o = gmax_ord[g * DPH + (j - DPH)];
        gv[j] = o ? o2f(o) : 0.0f;                   // empty-segment guard
    }
    __syncthreads();
    float acc = bp1[j];
#pragma unroll 8
    for (int i = 0; i < 2 * DPH; ++i) acc += gv[i] * Wp1[i * HID + j];
    acc = fmaxf(acc, 0.0f);                          // ReLU
    acc = (acc - rm[j]) * __frsqrt_rn(rv[j] + BN_EPS) * gamma[j] + beta[j];
    red[j] = acc * Wp2[j];                           // TASKS == 1
    __syncthreads();
    for (int stp = HID / 2; stp > 0; stp >>= 1) {
        if (j < stp) red[j] += red[j + stp];
        __syncthreads();
    }
    if (j == 0) out[g] = red[0] + bp2[0];
}

// ---------------------------------------------------------------------------
extern "C" void kernel_launch(void* const* d_in, const int* in_sizes, int n_in,
                              void* d_out, int out_size, void* d_ws, size_t ws_size,
                              hipStream_t stream)
{
    const float* x     = (const float*)d_in[0];
    const int*   src   = (const int*)  d_in[1];
    const int*   dst   = (const int*)  d_in[2];
    const int*   gid   = (const int*)  d_in[3];
    const float* W1    = (const float*)d_in[4];
    const float* al1   = (const float*)d_in[5];
    const float* ar1   = (const float*)d_in[6];
    const float* b1    = (const float*)d_in[7];
    const float* resW1 = (const float*)d_in[8];
    const float* W2    = (const float*)d_in[9];
    const float* al2   = (const float*)d_in[10];
    const float* ar2   = (const float*)d_in[11];
    const float* b2    = (const float*)d_in[12];
    const float* Ww    = (const float*)d_in[13];
    const float* bw    = (const float*)d_in[14];
    const float* Wp1   = (const float*)d_in[15];
    const float* bp1   = (const float*)d_in[16];
    const float* gamma = (const float*)d_in[17];
    const float* beta  = (const float*)d_in[18];
    const float* rm    = (const float*)d_in[19];
    const float* rv    = (const float*)d_in[20];
    const float* Wp2   = (const float*)d_in[21];
    const float* bp2   = (const float*)d_in[22];
    float* out = (float*)d_out;

    // ---- workspace layout (4-byte elements) ----
    float* ws = (float*)d_ws;
    const long NH  = (long)N_NODES * HD;     // 12.8M
    const long NHh = (long)N_NODES * HEADS;  // 400K
    const long EH  = (long)N_EDGES * HEADS;  // 6.4M
    const long NP  = (long)N_NODES * IN_PAD; // 1.6M
    float*    feat  = ws;                    // [N, 256]
    float*    agg   = feat + NH;             // [N, 256]
    float*    h1    = agg + NH;              // [N, 256]
    float*    el    = h1 + NH;               // [N, 8]
    float*    er    = el + NHh;              // [N, 8]
    unsigned* m_ord = (unsigned*)(er + NHh); // [N, 8]
    float*    ssum  = (float*)(m_ord + NHh); // [N, 8]
    float*    ebuf  = ssum + NHh;            // [E, 8]
    float*    gsum  = ebuf + EH;             // [G, 32]
    unsigned* gmax  = (unsigned*)(gsum + GRAPHS * DPH); // [G, 32]
    float*    xp    = (float*)(gmax + GRAPHS * DPH);    // [N, 32] padded x

    const int TB = 256;
    dim3 gemmGrid((M_TILES + 3) / 4, HD / 16);   // 4 M-tiles per block
    int nh_blocks  = (int)((NHh + TB - 1) / TB);
    int eh_blocks  = (int)((EH + TB - 1) / TB);
    int nhd_blocks = (int)((NH + TB - 1) / TB);

    // ======================= Layer 1 =======================
    k_pad<<<(int)((NP + TB - 1) / TB), TB, 0, stream>>>(x, xp, (int)NP);
    // feat = x @ W1 ; agg = x @ resW1 + b1   (residual+bias pre-folded)
    k_gemm_f32<IN_PAD><<<gemmGrid, 128, 0, stream>>>(xp, W1,    nullptr, feat, M_TILES, IN_F);
    k_gemm_f32<IN_PAD><<<gemmGrid, 128, 0, stream>>>(xp, resW1, b1,      agg,  M_TILES, IN_F);
    // zero segment max/sum state (m_ord + ssum are contiguous)
    k_zero<<<(int)((2 * NHh + TB - 1) / TB), TB, 0, stream>>>(m_ord, (int)(2 * NHh));
    k_elr<<<nh_blocks, TB, 0, stream>>>(feat, al1, ar1, el, er, (int)NHh);
    k_edge_max<<<eh_blocks, TB, 0, stream>>>(src, dst, el, er, ebuf, m_ord);
    k_edge_exp<<<eh_blocks, TB, 0, stream>>>(dst, ebuf, m_ord, ssum);
    k_edge_msg<<<N_EDGES, TB, 0, stream>>>(src, dst, ebuf, ssum, feat, agg);
    k_elu<<<nhd_blocks, TB, 0, stream>>>(agg, h1, (int)NH);

    // ======================= Layer 2 =======================
    // feat = h1 @ W2 ; agg = h1 + b2 (identity residual pre-folded)
    k_gemm_f32<HD><<<gemmGrid, 128, 0, stream>>>(h1, W2, nullptr, feat, M_TILES, HD);
    k_res2<<<nhd_blocks, TB, 0, stream>>>(h1, b2, agg, (int)NH);
    k_zero<<<(int)((2 * NHh + TB - 1) / TB), TB, 0, stream>>>(m_ord, (int)(2 * NHh));
    k_elr<<<nh_blocks, TB, 0, stream>>>(feat, al2, ar2, el, er, (int)NHh);
    k_edge_max<<<eh_blocks, TB, 0, stream>>>(src, dst, el, er, ebuf, m_ord);
    k_edge_exp<<<eh_blocks, TB, 0, stream>>>(dst, ebuf, m_ord, ssum);
    k_edge_msg<<<N_EDGES, TB, 0, stream>>>(src, dst, ebuf, ssum, feat, agg);

    // ======================= Readout + MLP =======================
    int gr_elems = GRAPHS * DPH * 2;  // gsum + gmax contiguous
    k_zero<<<(gr_elems + TB - 1) / TB, TB, 0, stream>>>((unsigned*)gsum, gr_elems);
    k_readout<<<(N_NODES + 7) / 8, TB, 0, stream>>>(agg, gid, Ww, bw, gsum, gmax);
    k_mlp<<<GRAPHS, HID, 0, stream>>>(gsum, gmax, Wp1, bp1, gamma, beta, rm, rv,
                                      Wp2, bp2, out);
}